// NodePredictionModel_9371618640687
// MI455X (gfx1250) — compile-verified
//
#include <hip/hip_runtime.h>
#include <math.h>

// MI455X / gfx1250: wave32, WMMA (V_WMMA_F32_16X16X4_F32) for all dense GEMMs.

typedef __attribute__((ext_vector_type(2))) float v2f;
typedef __attribute__((ext_vector_type(8))) float v8f;

__global__ void fill_kernel(float* __restrict__ p, float v, long n) {
    long i = (long)blockIdx.x * blockDim.x + threadIdx.x;
    if (i < n) p[i] = v;
}

__global__ void deg_kernel(const int* __restrict__ dst, float* __restrict__ deg, int E) {
    int i = blockIdx.x * blockDim.x + threadIdx.x;
    if (i < E) unsafeAtomicAdd(&deg[dst[i]], 1.0f);
}

__global__ void invden_kernel(float* __restrict__ deg, int N) {
    int i = blockIdx.x * blockDim.x + threadIdx.x;
    if (i < N) deg[i] = 1.0f / fmaxf(deg[i], 1.0f);
}

// Generic fp32 WMMA GEMM: Y[M,N] = act( X[M,K] @ W[N,K]^T + bias[N] )
// One wave computes a 16x64 output strip: 4 accumulators, 4 WMMAs per A-fragment
// load (4x less A traffic, 4x higher WMMA density per vmem op than 1 tile/wave).
// Requires M % 16 == 0, N % 64 == 0, K % 4 == 0 (true for every call here).
__global__ void wmma_gemm_xwT(const float* __restrict__ X, const float* __restrict__ W,
                              const float* __restrict__ bias, float* __restrict__ Y,
                              int M, int N, int K, int do_relu) {
    const int wavesPerBlock = blockDim.x >> 5;
    const int wid  = blockIdx.x * wavesPerBlock + (threadIdx.x >> 5);
    const int lane = threadIdx.x & 31;
    const int strips = N >> 6;                // N/64 strips per row-block
    const int mTile  = wid / strips;
    const int sTile  = wid - mTile * strips;
    if (mTile * 16 >= M) return;              // wave-uniform exit: EXEC all-1 for WMMA

    const int half = lane >> 4;               // 0: lanes 0-15, 1: lanes 16-31
    const int l16  = lane & 15;
    const int arow = mTile * 16 + l16;        // A-matrix row for this lane
    const int col0 = sTile * 64 + l16;        // first B/C column for this lane

    const float* __restrict__ Xr  = X + (size_t)arow * K;
    const float* __restrict__ Wr0 = W + (size_t)col0 * K;
    const float* __restrict__ Wr1 = Wr0 + (size_t)16 * K;
    const float* __restrict__ Wr2 = Wr0 + (size_t)32 * K;
    const float* __restrict__ Wr3 = Wr0 + (size_t)48 * K;

    v8f c0 = {}, c1 = {}, c2 = {}, c3 = {};
    for (int k = 0; k < K; k += 4) {
        // 32-bit 16x4 A layout: lanes 0-15 hold K=k,k+1; lanes 16-31 hold K=k+2,k+3
        const int kk = k + 2 * half;
        v2f a, b0, b1, b2, b3;
        a.x  = Xr[kk];   a.y  = Xr[kk + 1];
        b0.x = Wr0[kk];  b0.y = Wr0[kk + 1];
        b1.x = Wr1[kk];  b1.y = Wr1[kk + 1];
        b2.x = Wr2[kk];  b2.y = Wr2[kk + 1];
        b3.x = Wr3[kk];  b3.y = Wr3[kk + 1];
        c0 = __builtin_amdgcn_wmma_f32_16x16x4_f32(false, a, false, b0, (short)0, c0, false, false);
        c1 = __builtin_amdgcn_wmma_f32_16x16x4_f32(false, a, false, b1, (short)0, c1, false, false);
        c2 = __builtin_amdgcn_wmma_f32_16x16x4_f32(false, a, false, b2, (short)0, c2, false, false);
        c3 = __builtin_amdgcn_wmma_f32_16x16x4_f32(false, a, false, b3, (short)0, c3, false, false);
    }

    // C/D layout: VGPR r -> row r (lanes 0-15) / row r+8 (lanes 16-31), col = l16
    const size_t rowBase = (size_t)(mTile * 16 + 8 * half) * N;
    #pragma unroll
    for (int nt = 0; nt < 4; ++nt) {
        const int col = col0 + nt * 16;
        const float bv = bias ? bias[col] : 0.0f;
        const v8f& c = (nt == 0) ? c0 : (nt == 1) ? c1 : (nt == 2) ? c2 : c3;
        #pragma unroll
        for (int r = 0; r < 8; ++r) {
            float v = c[r] + bv;
            if (do_relu) v = fmaxf(v, 0.0f);
            Y[rowBase + (size_t)r * N + col] = v;
        }
    }
}

// Per-edge message: msg[e,o] = sum_i h[src[e],i] * ew[e, i*64+o], scatter-add to agg[dst].
// One wave per edge; lane owns output columns {lane, lane+32}; ew reads fully coalesced.
__global__ void msg_kernel(const int* __restrict__ src, const int* __restrict__ dst,
                           const float* __restrict__ h, const float* __restrict__ ew,
                           float* __restrict__ agg, int E) {
    const int wid = blockIdx.x * (blockDim.x >> 5) + (threadIdx.x >> 5);
    if (wid >= E) return;
    const int lane = threadIdx.x & 31;
    const int s = src[wid], d = dst[wid];
    const float* __restrict__ We = ew + (size_t)wid * 4096;
    const float* __restrict__ xs = h + (size_t)s * 64;
    float a0 = 0.0f, a1 = 0.0f;
    #pragma unroll 4
    for (int i = 0; i < 64; ++i) {
        const float xv = xs[i];                       // uniform across wave (cached)
        a0 = fmaf(xv, We[i * 64 + lane], a0);
        a1 = fmaf(xv, We[i * 64 + lane + 32], a1);
    }
    unsafeAtomicAdd(&agg[(size_t)d * 64 + lane], a0);
    unsafeAtomicAdd(&agg[(size_t)d * 64 + lane + 32], a1);
}

// m = relu(agg * invden[row] + conv)   (conv already contains bconv)
__global__ void combine_kernel(const float* __restrict__ agg, const float* __restrict__ invden,
                               const float* __restrict__ conv, float* __restrict__ m, long n) {
    long i = (long)blockIdx.x * blockDim.x + threadIdx.x;
    if (i < n) m[i] = fmaxf(agg[i] * invden[i >> 6] + conv[i], 0.0f);
}

__device__ __forceinline__ float sigm(float x) { return 1.0f / (1.0f + __expf(-x)); }

// One GRU step (torch gate order r,z,n); updates h in place.
__global__ void gru_kernel(const float* __restrict__ gi, const float* __restrict__ gh,
                           float* __restrict__ h, int N) {
    int idx = blockIdx.x * blockDim.x + threadIdx.x;
    if (idx >= N * 64) return;
    const int n = idx >> 6, d = idx & 63;
    const float* gin = gi + (size_t)n * 192;
    const float* ghn = gh + (size_t)n * 192;
    const float r  = sigm(gin[d] + ghn[d]);
    const float z  = sigm(gin[64 + d] + ghn[64 + d]);
    const float nn = tanhf(gin[128 + d] + r * ghn[128 + d]);
    h[idx] = (1.0f - z) * nn + z * h[idx];
}

// y[n] = relu(cat(h[n], gattr[batch[n]]) @ W1^T + b1) @ W2^T + b2  (K=72, OUT=1)
// One wave per node; lane owns hidden rows {lane, lane+32}; shuffle reduction for the dot.
__global__ void final_kernel(const float* __restrict__ h, const int* __restrict__ batch,
                             const float* __restrict__ gattr,
                             const float* __restrict__ W1, const float* __restrict__ b1,
                             const float* __restrict__ W2, const float* __restrict__ b2,
                             float* __restrict__ y, int N) {
    const int wid = blockIdx.x * (blockDim.x >> 5) + (threadIdx.x >> 5);
    if (wid >= N) return;
    const int lane = threadIdx.x & 31;
    const float* __restrict__ hn = h + (size_t)wid * 64;
    const int g = batch[wid];
    const float* __restrict__ ga = gattr + (size_t)g * 8;
    float h0 = b1[lane], h1 = b1[lane + 32];
    const float* w1a = W1 + (size_t)lane * 72;
    const float* w1b = W1 + (size_t)(lane + 32) * 72;
    #pragma unroll 8
    for (int k = 0; k < 64; ++k) {
        const float cv = hn[k];
        h0 = fmaf(cv, w1a[k], h0);
        h1 = fmaf(cv, w1b[k], h1);
    }
    #pragma unroll
    for (int k = 0; k < 8; ++k) {
        const float cv = ga[k];
        h0 = fmaf(cv, w1a[64 + k], h0);
        h1 = fmaf(cv, w1b[64 + k], h1);
    }
    h0 = fmaxf(h0, 0.0f);
    h1 = fmaxf(h1, 0.0f);
    float p = h0 * W2[lane] + h1 * W2[lane + 32];
    #pragma unroll
    for (int off = 16; off > 0; off >>= 1) p += __shfl_down(p, off, 32);
    if (lane == 0) y[wid] = p + b2[0];
}

static inline int gemm_blocks(int M, int N) {   // 8 waves (16x64 strips) per 256-thread block
    const int strips = (M / 16) * (N / 64);
    return (strips + 7) / 8;
}

extern "C" void kernel_launch(void* const* d_in, const int* in_sizes, int n_in,
                              void* d_out, int out_size, void* d_ws, size_t ws_size,
                              hipStream_t stream) {
    const float* x     = (const float*)d_in[0];
    const int*   eidx  = (const int*)  d_in[1];
    const float* eattr = (const float*)d_in[2];
    const int*   batch = (const int*)  d_in[3];
    const float* gattr = (const float*)d_in[4];
    const float* W0    = (const float*)d_in[5];
    const float* b0    = (const float*)d_in[6];
    const float* We1   = (const float*)d_in[7];
    const float* be1   = (const float*)d_in[8];
    const float* We2   = (const float*)d_in[9];
    const float* be2   = (const float*)d_in[10];
    const float* Wroot = (const float*)d_in[11];
    const float* bconv = (const float*)d_in[12];
    const float* W_ih  = (const float*)d_in[13];
    const float* W_hh  = (const float*)d_in[14];
    const float* b_ih  = (const float*)d_in[15];
    const float* b_hh  = (const float*)d_in[16];
    const float* W1    = (const float*)d_in[17];
    const float* b1    = (const float*)d_in[18];
    const float* W2    = (const float*)d_in[19];
    const float* b2    = (const float*)d_in[20];

    const int N = in_sizes[3];        // batch: [N]
    const int E = in_sizes[1] / 2;    // edge_index: [2,E]
    const int* src = eidx;
    const int* dst = eidx + E;

    // Workspace carve-up (fp32)
    float* ws = (float*)d_ws;
    size_t off = 0;
    float* ew   = ws + off; off += (size_t)E * 4096;   // per-edge 64x64 weights (loop-invariant)
    float* h1   = ws + off; off += (size_t)E * 128;    // edge MLP hidden
    float* hbuf = ws + off; off += (size_t)N * 64;     // out == h (identical in ref loop)
    float* conv = ws + off; off += (size_t)N * 64;     // out @ Wroot^T + bconv
    float* agg  = ws + off; off += (size_t)N * 64;     // scatter-mean accumulator
    float* mbuf = ws + off; off += (size_t)N * 64;
    float* gi   = ws + off; off += (size_t)N * 192;
    float* gh   = ws + off; off += (size_t)N * 192;
    float* deg  = ws + off; off += (size_t)N;          // becomes invden in place
    if (ws_size < off * sizeof(float)) return;         // need ~870 MB scratch

    const int T = 256;

    // Degree -> 1/max(deg,1)
    fill_kernel<<<(N + T - 1) / T, T, 0, stream>>>(deg, 0.0f, (long)N);
    deg_kernel<<<(E + T - 1) / T, T, 0, stream>>>(dst, deg, E);
    invden_kernel<<<(N + T - 1) / T, T, 0, stream>>>(deg, N);

    // Edge MLP (loop-invariant): h1 = relu(eattr @ We1^T + be1); ew = h1 @ We2^T + be2
    wmma_gemm_xwT<<<gemm_blocks(E, 128), T, 0, stream>>>(eattr, We1, be1, h1, E, 128, 16, 1);
    wmma_gemm_xwT<<<gemm_blocks(E, 4096), T, 0, stream>>>(h1, We2, be2, ew, E, 4096, 128, 0);

    // lin0: h = relu(x @ W0^T + b0)
    wmma_gemm_xwT<<<gemm_blocks(N, 64), T, 0, stream>>>(x, W0, b0, hbuf, N, 64, 32, 1);

    for (int t = 0; t < 3; ++t) {
        fill_kernel<<<((long)N * 64 + T - 1) / T, T, 0, stream>>>(agg, 0.0f, (long)N * 64);
        msg_kernel<<<(E + 7) / 8, T, 0, stream>>>(src, dst, hbuf, ew, agg, E);
        wmma_gemm_xwT<<<gemm_blocks(N, 64), T, 0, stream>>>(hbuf, Wroot, bconv, conv, N, 64, 64, 0);
        combine_kernel<<<((long)N * 64 + T - 1) / T, T, 0, stream>>>(agg, deg, conv, mbuf, (long)N * 64);
        wmma_gemm_xwT<<<gemm_blocks(N, 192), T, 0, stream>>>(mbuf, W_ih, b_ih, gi, N, 192, 64, 0);
        wmma_gemm_xwT<<<gemm_blocks(N, 192), T, 0, stream>>>(hbuf, W_hh, b_hh, gh, N, 192, 64, 0);
        gru_kernel<<<(N * 64 + T - 1) / T, T, 0, stream>>>(gi, gh, hbuf, N);
    }

    final_kernel<<<(N + 7) / 8, T, 0, stream>>>(hbuf, batch, gattr, W1, b1, W2, b2,
                                                (float*)d_out, N);
}